// layer_83794811945447
// MI455X (gfx1250) — compile-verified
//
#include <hip/hip_runtime.h>
#include <hip/hip_bf16.h>

typedef __attribute__((ext_vector_type(2))) float v2f;
typedef __attribute__((ext_vector_type(8))) float v8f;

#define F_IN 128
#define HN 4
#define F_OUT 64
#define HF 256   // H * F_OUT

// ---------------------------------------------------------------------------
// float atomic max via int trick (init value must be -inf = 0xFF800000)
// ---------------------------------------------------------------------------
__device__ inline void atomicMaxF32(float* addr, float val) {
    if (val >= 0.0f) {
        atomicMax((int*)addr, __float_as_int(val));
    } else {
        atomicMin((unsigned int*)addr, (unsigned int)__float_as_int(val));
    }
}

// ---------------------------------------------------------------------------
// Kernel 0: init out = bias (broadcast), m = -inf, denom = 0
// ---------------------------------------------------------------------------
__global__ void gat_init(float* __restrict__ out, const float* __restrict__ bias,
                         float* __restrict__ m, float* __restrict__ denom, int n_nodes) {
    int i = blockIdx.x * blockDim.x + threadIdx.x;
    int total = n_nodes * HF;
    if (i < total) out[i] = bias[i & (HF - 1)];
    if (i < n_nodes * HN) {
        m[i] = -__builtin_inff();
        denom[i] = 0.0f;
    }
}

// ---------------------------------------------------------------------------
// Kernel 1: per-node quadratic terms
//   el2[n,h] = sum_f feat[n,f]^2 * attn_l1[h,f]^2 ; er2 analogous
// one wave per node, float4 loads, xor-shuffle reduce
// ---------------------------------------------------------------------------
__global__ void gat_node_norm(const float* __restrict__ feat,
                              const float* __restrict__ attn_l1,
                              const float* __restrict__ attn_r1,
                              float* __restrict__ el2, float* __restrict__ er2,
                              int n_nodes) {
    __shared__ float wl[HN][F_IN];
    __shared__ float wr[HN][F_IN];
    for (int i = threadIdx.x; i < HN * F_IN; i += blockDim.x) {
        float l = attn_l1[i], r = attn_r1[i];
        wl[i >> 7][i & 127] = l * l;
        wr[i >> 7][i & 127] = r * r;
    }
    __syncthreads();

    int lane = threadIdx.x & 31;
    int node = blockIdx.x * (blockDim.x >> 5) + (threadIdx.x >> 5);
    if (node >= n_nodes) return;

    float4 a = ((const float4*)(feat + (size_t)node * F_IN))[lane];
    float q0 = a.x * a.x, q1 = a.y * a.y, q2 = a.z * a.z, q3 = a.w * a.w;
    int f = lane * 4;

    float al[HN], ar[HN];
#pragma unroll
    for (int h = 0; h < HN; ++h) {
        al[h] = q0 * wl[h][f] + q1 * wl[h][f + 1] + q2 * wl[h][f + 2] + q3 * wl[h][f + 3];
        ar[h] = q0 * wr[h][f] + q1 * wr[h][f + 1] + q2 * wr[h][f + 2] + q3 * wr[h][f + 3];
    }
#pragma unroll
    for (int off = 16; off >= 1; off >>= 1) {
#pragma unroll
        for (int h = 0; h < HN; ++h) {
            al[h] += __shfl_xor(al[h], off, 32);
            ar[h] += __shfl_xor(ar[h], off, 32);
        }
    }
    if (lane < HN) {
        el2[(size_t)node * HN + lane] = al[lane];
        er2[(size_t)node * HN + lane] = ar[lane];
    }
}

// ---------------------------------------------------------------------------
// Kernel 2: ft = feat @ fc_w.T   ([N,128] x [128,256]) via V_WMMA_F32_16X16X4_F32
// One wave computes one 16x16 output tile; K=128 -> 32 wmma steps.
// A (16x4 f32): lanes 0-15 hold K=0,1 in v[0],v[1]; lanes 16-31 hold K=2,3.
// B (4x16 f32): lanes 0-15 hold rows K=0,1 (col = lane); lanes 16-31 rows K=2,3.
// C/D (16x16 f32): VGPR r = row (r + 8*(lane>>4)), col = lane&15.
// ---------------------------------------------------------------------------
__global__ void gat_gemm_wmma(const float* __restrict__ feat,
                              const float* __restrict__ fc_w,
                              float* __restrict__ ft, int n_nodes) {
    int gwave = (blockIdx.x * blockDim.x + threadIdx.x) >> 5;
    int lane = threadIdx.x & 31;
    const int jt_tiles = HF / 16;             // 16 tiles over 256 cols
    int mt = gwave / jt_tiles;
    int jt = gwave % jt_tiles;
    int m_tiles = (n_nodes + 15) >> 4;
    if (mt >= m_tiles) return;

    int row = mt * 16 + (lane & 15);
    if (row >= n_nodes) row = n_nodes - 1;     // clamp (N=50000 divides 16 anyway)
    int col = jt * 16 + (lane & 15);           // output column = fc_w row
    int ksel = (lane >> 4) * 2;                // 0 or 2

    const float* arow = feat + (size_t)row * F_IN + ksel;
    const float* brow = fc_w + (size_t)col * F_IN + ksel;  // B[k][col] = fc_w[col][k]

    v8f c = {};
#pragma unroll
    for (int k = 0; k < F_IN; k += 4) {
        v2f a = *(const v2f*)(arow + k);
        v2f b = *(const v2f*)(brow + k);
        c = __builtin_amdgcn_wmma_f32_16x16x4_f32(false, a, false, b,
                                                  (short)0, c, false, false);
    }

    int mbase = mt * 16 + ((lane >> 4) << 3);
    int n = jt * 16 + (lane & 15);
#pragma unroll
    for (int r = 0; r < 8; ++r) {
        int mrow = mbase + r;
        if (mrow < n_nodes)
            ft[(size_t)mrow * HF + n] = c[r];
    }
}

// ---------------------------------------------------------------------------
// Kernel 3: edge scores
//   e[e,h] = sum_f fs*fd*wc[h,f] + el2[src,h] + er2[dst,h],  wc = al*ar - 2*al1*ar1
//   and atomic segment-max into m[dst,h]
// one wave per edge; coalesced float4 gathers of both endpoint rows
// ---------------------------------------------------------------------------
__global__ void gat_edge_scores(const float* __restrict__ feat,
                                const int* __restrict__ src, const int* __restrict__ dst,
                                const float* __restrict__ attn_l, const float* __restrict__ attn_r,
                                const float* __restrict__ attn_l1, const float* __restrict__ attn_r1,
                                const float* __restrict__ el2, const float* __restrict__ er2,
                                float* __restrict__ e_out, float* __restrict__ m, int n_edges) {
    __shared__ float wc[HN][F_IN];
    for (int i = threadIdx.x; i < HN * F_IN; i += blockDim.x) {
        wc[i >> 7][i & 127] = attn_l[i] * attn_r[i] - 2.0f * attn_l1[i] * attn_r1[i];
    }
    __syncthreads();

    int lane = threadIdx.x & 31;
    int e = blockIdx.x * (blockDim.x >> 5) + (threadIdx.x >> 5);
    if (e >= n_edges) return;
    int s = src[e], d = dst[e];

    float4 a = ((const float4*)(feat + (size_t)s * F_IN))[lane];
    float4 b = ((const float4*)(feat + (size_t)d * F_IN))[lane];
    float p0 = a.x * b.x, p1 = a.y * b.y, p2 = a.z * b.z, p3 = a.w * b.w;
    int f = lane * 4;

    float acc[HN];
#pragma unroll
    for (int h = 0; h < HN; ++h)
        acc[h] = p0 * wc[h][f] + p1 * wc[h][f + 1] + p2 * wc[h][f + 2] + p3 * wc[h][f + 3];

#pragma unroll
    for (int off = 16; off >= 1; off >>= 1)
#pragma unroll
        for (int h = 0; h < HN; ++h)
            acc[h] += __shfl_xor(acc[h], off, 32);

    if (lane < HN) {
        int h = lane;
        float val = acc[h] + el2[(size_t)s * HN + h] + er2[(size_t)d * HN + h];
        e_out[(size_t)e * HN + h] = val;
        atomicMaxF32(&m[(size_t)d * HN + h], val);
    }
}

// ---------------------------------------------------------------------------
// Kernel 4: ex = exp(e - m[dst]); denom[dst] += ex   (in-place over e buffer)
// one thread per edge (4 heads as float4)
// ---------------------------------------------------------------------------
__global__ void gat_edge_exp(const float* e_in, const int* __restrict__ dst,
                             const float* __restrict__ m,
                             float* ex, float* __restrict__ denom, int n_edges) {
    int e = blockIdx.x * blockDim.x + threadIdx.x;
    if (e >= n_edges) return;
    int d = dst[e];
    float4 ev = ((const float4*)e_in)[e];
    float4 mv = ((const float4*)m)[d];
    float x0 = __expf(ev.x - mv.x);
    float x1 = __expf(ev.y - mv.y);
    float x2 = __expf(ev.z - mv.z);
    float x3 = __expf(ev.w - mv.w);
    ((float4*)ex)[e] = make_float4(x0, x1, x2, x3);
    float* dn = denom + (size_t)d * HN;
    atomicAdd(dn + 0, x0);
    atomicAdd(dn + 1, x1);
    atomicAdd(dn + 2, x2);
    atomicAdd(dn + 3, x3);
}

// ---------------------------------------------------------------------------
// Kernel 5: out[dst, h, f] += ft[src, h, f] * (ex[e,h] / denom[dst,h])
// one wave per edge; each lane handles 8 contiguous outputs (same head)
// ---------------------------------------------------------------------------
__global__ void gat_aggregate(const float* __restrict__ ex,
                              const int* __restrict__ src, const int* __restrict__ dst,
                              const float* __restrict__ denom,
                              const float* __restrict__ ft,
                              float* __restrict__ out, int n_edges) {
    int lane = threadIdx.x & 31;
    int e = blockIdx.x * (blockDim.x >> 5) + (threadIdx.x >> 5);
    if (e >= n_edges) return;
    int s = src[e], d = dst[e];

    float4 exv = ((const float4*)ex)[e];
    float4 dnv = ((const float4*)denom)[d];
    float av[HN] = {exv.x / dnv.x, exv.y / dnv.y, exv.z / dnv.z, exv.w / dnv.w};

    int j = lane * 8;             // 8 contiguous of 256; all in head j/64
    float a = av[j >> 6];
    const float4* ftp = (const float4*)(ft + (size_t)s * HF + j);
    float4 u = ftp[0], v = ftp[1];
    float* op = out + (size_t)d * HF + j;
    atomicAdd(op + 0, u.x * a);
    atomicAdd(op + 1, u.y * a);
    atomicAdd(op + 2, u.z * a);
    atomicAdd(op + 3, u.w * a);
    atomicAdd(op + 4, v.x * a);
    atomicAdd(op + 5, v.y * a);
    atomicAdd(op + 6, v.z * a);
    atomicAdd(op + 7, v.w * a);
}

// ---------------------------------------------------------------------------
extern "C" void kernel_launch(void* const* d_in, const int* in_sizes, int n_in,
                              void* d_out, int out_size, void* d_ws, size_t ws_size,
                              hipStream_t stream) {
    const float* feat    = (const float*)d_in[0];
    const int*   src     = (const int*)  d_in[1];
    const int*   dst     = (const int*)  d_in[2];
    const float* fc_w    = (const float*)d_in[3];
    const float* attn_l  = (const float*)d_in[4];
    const float* attn_r  = (const float*)d_in[5];
    const float* attn_l1 = (const float*)d_in[6];
    const float* attn_r1 = (const float*)d_in[7];
    const float* bias    = (const float*)d_in[8];
    float* out = (float*)d_out;

    const int N = in_sizes[0] / F_IN;
    const int E = in_sizes[1];

    // workspace carve-out (256B aligned slabs)
    char* ws = (char*)d_ws;
    size_t off = 0;
    auto carve = [&](size_t bytes) -> void* {
        void* p = ws + off;
        off += (bytes + 255) & ~(size_t)255;
        return p;
    };
    float* ft    = (float*)carve((size_t)N * HF * sizeof(float));   // 51.2 MB
    float* ebuf  = (float*)carve((size_t)E * HN * sizeof(float));   // 12.8 MB (e, then ex in-place)
    float* m     = (float*)carve((size_t)N * HN * sizeof(float));
    float* denom = (float*)carve((size_t)N * HN * sizeof(float));
    float* el2   = (float*)carve((size_t)N * HN * sizeof(float));
    float* er2   = (float*)carve((size_t)N * HN * sizeof(float));
    (void)ws_size;

    const int BT = 256;                 // 8 waves / block
    // 0) init out with bias, m with -inf, denom with 0
    {
        int total = N * HF;
        gat_init<<<(total + BT - 1) / BT, BT, 0, stream>>>(out, bias, m, denom, N);
    }
    // 1) per-node quadratic terms
    gat_node_norm<<<(N + 7) / 8, BT, 0, stream>>>(feat, attn_l1, attn_r1, el2, er2, N);
    // 2) WMMA GEMM: ft = feat @ fc_w.T
    {
        int m_tiles = (N + 15) / 16;
        int waves = m_tiles * (HF / 16);
        gat_gemm_wmma<<<(waves + 7) / 8, BT, 0, stream>>>(feat, fc_w, ft, N);
    }
    // 3) edge scores + segment max
    gat_edge_scores<<<(E + 7) / 8, BT, 0, stream>>>(feat, src, dst, attn_l, attn_r,
                                                    attn_l1, attn_r1, el2, er2,
                                                    ebuf, m, E);
    // 4) exp + segment sum
    gat_edge_exp<<<(E + BT - 1) / BT, BT, 0, stream>>>(ebuf, dst, m, ebuf, denom, E);
    // 5) weighted aggregation into out (pre-seeded with bias)
    gat_aggregate<<<(E + 7) / 8, BT, 0, stream>>>(ebuf, src, dst, denom, ft, out, E);
}